// DeepEdgeFeatureGAT_9706626089367
// MI455X (gfx1250) — compile-verified
//
#include <hip/hip_runtime.h>
#include <math.h>

#define NN  100000
#define EE  1000000
#define DD  128
#define EDD 16
#define LL  5
#define MIDN 3
#define ETOT (EE + NN)

typedef float v2f __attribute__((ext_vector_type(2)));
typedef float v8f __attribute__((ext_vector_type(8)));

// ---------------------------------------------------------------------------
// float atomic max via monotonic int/uint trick (init must be -inf)
// ---------------------------------------------------------------------------
__device__ __forceinline__ void atomicMaxF(float* addr, float v) {
  if (v >= 0.0f) atomicMax((int*)addr, __float_as_int(v));
  else           atomicMin((unsigned int*)addr, __float_as_uint(v));
}

// ---------------------------------------------------------------------------
// C[M x 128] = A[M x 128] * B[128 x 128], exact f32 via V_WMMA_F32_16X16X4_F32.
// grid.x = M/16 row tiles; block = 256 threads = 8 waves; wave w owns the
// 16x16 output tile at columns [16w, 16w+16).
// A 16x4 f32 layout: lanes 0-15 = M rows, VGPR0 = K={0,2}, VGPR1 = K={1,3}.
// B 4x16 f32 layout: lanes 0-15 = N cols, VGPR0 = K={0,2}, VGPR1 = K={1,3}.
// C 16x16 f32 layout: VGPR r -> row r (lanes 0-15) / row r+8 (lanes 16-31).
// ---------------------------------------------------------------------------
__global__ void gemm128_wmma_f32(const float* __restrict__ A,
                                 const float* __restrict__ B,
                                 float* __restrict__ C) {
  const int tm   = blockIdx.x;          // 16-row tile
  const int tn   = threadIdx.x >> 5;    // 16-col tile (0..7)
  const int lane = threadIdx.x & 31;
  const int half = lane >> 4;           // 0 or 1
  const int l16  = lane & 15;
  const float* arow = A + (size_t)(tm * 16 + l16) * DD;
  const float* bcol = B + tn * 16 + l16;
  v8f acc = {};
#pragma unroll 4
  for (int k = 0; k < DD; k += 4) {
    const int ka = k + 2 * half;
    v2f a, b;
    a.x = arow[ka];
    a.y = arow[ka + 1];
    b.x = bcol[(size_t)ka * DD];
    b.y = bcol[(size_t)(ka + 1) * DD];
    acc = __builtin_amdgcn_wmma_f32_16x16x4_f32(false, a, false, b,
                                                (short)0, acc, false, false);
  }
#pragma unroll
  for (int r = 0; r < 8; ++r) {
    C[(size_t)(tm * 16 + r + 8 * half) * DD + tn * 16 + l16] = acc[r];
  }
}

// ---------------------------------------------------------------------------
// generic fill
// ---------------------------------------------------------------------------
__global__ void fill_f32(float* p, float v, int n) {
  int i = blockIdx.x * blockDim.x + threadIdx.x;
  if (i < n) p[i] = v;
}

// agg[i][j] = bias[j]  (folds the +b into the aggregation buffer)
__global__ void init_bias_kernel(float* __restrict__ agg, const float* __restrict__ b) {
  int i = blockIdx.x * blockDim.x + threadIdx.x;
  if (i >= NN * DD) return;
  agg[i] = b[i & (DD - 1)];
}

// ---------------------------------------------------------------------------
// self-loop attr precompute: deg + sum of incoming edge_attr
// ---------------------------------------------------------------------------
__global__ void deg_loop_kernel(const int* __restrict__ ei, const float* __restrict__ ea,
                                float* __restrict__ deg, float* __restrict__ lacc) {
  int e = blockIdx.x * blockDim.x + threadIdx.x;
  if (e >= EE) return;
  int d = ei[EE + e];
  atomicAdd(deg + d, 1.0f);
  const float* row = ea + (size_t)e * EDD;
  float* o = lacc + (size_t)d * EDD;
#pragma unroll
  for (int k = 0; k < EDD; ++k) atomicAdd(o + k, row[k]);
}

__global__ void loop_final_kernel(float* __restrict__ lacc, const float* __restrict__ deg) {
  int i = blockIdx.x * blockDim.x + threadIdx.x;
  if (i >= NN * EDD) return;
  lacc[i] /= fmaxf(deg[i / EDD], 1.0f);
}

// we_ae[l][k] = sum_j Wes[l][k][j] * a_edges[l][j]   (5*16 values)
__global__ void weae_kernel(const float* __restrict__ Wes, const float* __restrict__ a_edges,
                            float* __restrict__ weae) {
  int t = threadIdx.x;
  if (t >= LL * EDD) return;
  int l = t / EDD, k = t % EDD;
  const float* wr  = Wes + ((size_t)l * EDD + k) * DD;
  const float* aev = a_edges + (size_t)l * DD;
  float acc = 0.f;
  for (int j = 0; j < DD; ++j) acc += wr[j] * aev[j];
  weae[t] = acc;
}

// ---------------------------------------------------------------------------
// hs[i] = h[i] . a_src ; hd[i] = h[i] . a_dst   (one wave32 per node)
// ---------------------------------------------------------------------------
__global__ void rowdot_kernel(const float* __restrict__ h, const float* __restrict__ as,
                              const float* __restrict__ ad,
                              float* __restrict__ hs, float* __restrict__ hd) {
  int node = blockIdx.x * 8 + (threadIdx.x >> 5);
  int lane = threadIdx.x & 31;
  if (node >= NN) return;
  float4 v = ((const float4*)(h + (size_t)node * DD))[lane];
  float4 a = ((const float4*)as)[lane];
  float4 b = ((const float4*)ad)[lane];
  float ds = v.x * a.x + v.y * a.y + v.z * a.z + v.w * a.w;
  float dd = v.x * b.x + v.y * b.y + v.z * b.z + v.w * b.w;
#pragma unroll
  for (int off = 16; off > 0; off >>= 1) {
    ds += __shfl_xor(ds, off, 32);
    dd += __shfl_xor(dd, off, 32);
  }
  if (lane == 0) { hs[node] = ds; hd[node] = dd; }
}

// ---------------------------------------------------------------------------
// edge pass 1: raw attention logit (leaky 0.2) + segment max over dst
// edges [0,EE) real, [EE, ETOT) implicit self loops
// ---------------------------------------------------------------------------
__global__ void edge_logits_kernel(const float* __restrict__ hs, const float* __restrict__ hd,
                                   const float* __restrict__ ea, const float* __restrict__ lacc,
                                   const float* __restrict__ weae_l, const int* __restrict__ ei,
                                   float* __restrict__ sedg, float* __restrict__ smax) {
  int e = blockIdx.x * blockDim.x + threadIdx.x;
  if (e >= ETOT) return;
  int s, d;
  const float* arow;
  if (e < EE) { s = ei[e]; d = ei[EE + e]; arow = ea + (size_t)e * EDD; }
  else        { s = d = e - EE;            arow = lacc + (size_t)(e - EE) * EDD; }
  float acc = 0.f;
#pragma unroll
  for (int k = 0; k < EDD; ++k) acc += arow[k] * weae_l[k];
  float v = hs[s] + hd[d] + acc;
  v = (v > 0.f) ? v : 0.2f * v;
  sedg[e] = v;
  atomicMaxF(smax + d, v);
}

// edge pass 2: ex = exp(s - smax[dst]); segment sum -> denom
__global__ void edge_exp_kernel(const int* __restrict__ ei, float* __restrict__ sedg,
                                const float* __restrict__ smax, float* __restrict__ denom) {
  int e = blockIdx.x * blockDim.x + threadIdx.x;
  if (e >= ETOT) return;
  int d = (e < EE) ? ei[EE + e] : (e - EE);
  float ex = expf(sedg[e] - smax[d]);
  sedg[e] = ex;
  atomicAdd(denom + d, ex);
}

// edge pass 3: out[dst] += (ex/denom[dst]) * h[src]   (one wave32 per edge)
__global__ void edge_agg_kernel(const float* __restrict__ h, const float* __restrict__ sedg,
                                const float* __restrict__ denom, const int* __restrict__ ei,
                                float* __restrict__ agg) {
  int e    = blockIdx.x * 8 + (threadIdx.x >> 5);
  int lane = threadIdx.x & 31;
  if (e >= ETOT) return;
  int s, d;
  if (e < EE) { s = ei[e]; d = ei[EE + e]; } else { s = d = e - EE; }
  float alpha = sedg[e] / denom[d];
  float4 v = ((const float4*)(h + (size_t)s * DD))[lane];
  float* o = agg + (size_t)d * DD + lane * 4;
  atomicAdd(o + 0, alpha * v.x);
  atomicAdd(o + 1, alpha * v.y);
  atomicAdd(o + 2, alpha * v.z);
  atomicAdd(o + 3, alpha * v.w);
}

// ---------------------------------------------------------------------------
// GCNII-style mid-layer combines
// ---------------------------------------------------------------------------
__global__ void combine1_kernel(const float* __restrict__ c, const float* __restrict__ x0,
                                float* __restrict__ t) {
  int i = blockIdx.x * blockDim.x + threadIdx.x;
  if (i >= NN * DD) return;
  t[i] = 0.8f * c[i] + 0.2f * x0[i];
}

__global__ void combine2_kernel(float* __restrict__ t, const float* __restrict__ g, float beta) {
  int i = blockIdx.x * blockDim.x + threadIdx.x;
  if (i >= NN * DD) return;
  float v = (1.0f - beta) * t[i] + beta * g[i];
  t[i] = (v > 0.f) ? v : 0.01f * v;
}

// ---------------------------------------------------------------------------
extern "C" void kernel_launch(void* const* d_in, const int* in_sizes, int n_in,
                              void* d_out, int out_size, void* d_ws, size_t ws_size,
                              hipStream_t stream) {
  const float* x       = (const float*)d_in[0];
  const int*   ei      = (const int*)d_in[1];
  const float* ea      = (const float*)d_in[2];
  const float* Ws      = (const float*)d_in[3];
  const float* a_srcs  = (const float*)d_in[5];
  const float* a_dsts  = (const float*)d_in[6];
  const float* a_edges = (const float*)d_in[7];
  const float* Wes     = (const float*)d_in[4];
  const float* biases  = (const float*)d_in[8];
  const float* mid_ws  = (const float*)d_in[9];
  float* out = (float*)d_out;

  float* w = (float*)d_ws;
  float* HB    = w; w += (size_t)NN * DD;   // h = x @ W scratch
  float* X0    = w; w += (size_t)NN * DD;   // first layer output
  float* C1    = w; w += (size_t)NN * DD;   // gat output / mid gemm output
  float* T     = w; w += (size_t)NN * DD;   // running hidden state
  float* hsv   = w; w += NN;
  float* hdv   = w; w += NN;
  float* smax  = w; w += NN;
  float* denom = w; w += NN;
  float* deg   = w; w += NN;
  float* lacc  = w; w += (size_t)NN * EDD;  // self-loop attr
  float* sedg  = w; w += ETOT;              // per-edge logit / exp
  float* weae  = w; w += LL * EDD;          // collapsed edge weights

  const int TPB = 256;
  // ---- self-loop attrs + collapsed edge projections (once) ----
  fill_f32<<<dim3((NN + TPB - 1) / TPB), dim3(TPB), 0, stream>>>(deg, 0.f, NN);
  fill_f32<<<dim3((NN * EDD + TPB - 1) / TPB), dim3(TPB), 0, stream>>>(lacc, 0.f, NN * EDD);
  deg_loop_kernel<<<dim3((EE + TPB - 1) / TPB), dim3(TPB), 0, stream>>>(ei, ea, deg, lacc);
  loop_final_kernel<<<dim3((NN * EDD + TPB - 1) / TPB), dim3(TPB), 0, stream>>>(lacc, deg);
  weae_kernel<<<dim3(1), dim3(128), 0, stream>>>(Wes, a_edges, weae);

  auto launch_gat = [&](const float* xin, int l, float* aggout) {
    gemm128_wmma_f32<<<dim3(NN / 16), dim3(TPB), 0, stream>>>(
        xin, Ws + (size_t)l * DD * DD, HB);
    rowdot_kernel<<<dim3(NN / 8), dim3(TPB), 0, stream>>>(
        HB, a_srcs + (size_t)l * DD, a_dsts + (size_t)l * DD, hsv, hdv);
    fill_f32<<<dim3((NN + TPB - 1) / TPB), dim3(TPB), 0, stream>>>(smax, -INFINITY, NN);
    fill_f32<<<dim3((NN + TPB - 1) / TPB), dim3(TPB), 0, stream>>>(denom, 0.f, NN);
    init_bias_kernel<<<dim3(NN * DD / TPB), dim3(TPB), 0, stream>>>(
        aggout, biases + (size_t)l * DD);
    edge_logits_kernel<<<dim3((ETOT + TPB - 1) / TPB), dim3(TPB), 0, stream>>>(
        hsv, hdv, ea, lacc, weae + l * EDD, ei, sedg, smax);
    edge_exp_kernel<<<dim3((ETOT + TPB - 1) / TPB), dim3(TPB), 0, stream>>>(
        ei, sedg, smax, denom);
    edge_agg_kernel<<<dim3(ETOT / 8), dim3(TPB), 0, stream>>>(
        HB, sedg, denom, ei, aggout);
  };

  // ---- layer 0 ----
  launch_gat(x, 0, X0);
  const float* hcur = X0;
  // ---- GCNII mid layers ----
  for (int i = 0; i < MIDN; ++i) {
    launch_gat(hcur, i + 1, C1);
    combine1_kernel<<<dim3(NN * DD / TPB), dim3(TPB), 0, stream>>>(C1, X0, T);
    gemm128_wmma_f32<<<dim3(NN / 16), dim3(TPB), 0, stream>>>(
        T, mid_ws + (size_t)i * DD * DD, C1);
    float beta = logf(0.2f / (float)(i + 1) + 1.0f);
    combine2_kernel<<<dim3(NN * DD / TPB), dim3(TPB), 0, stream>>>(T, C1, beta);
    hcur = T;
  }
  // ---- final layer straight into d_out ----
  launch_gat(hcur, LL - 1, out);
}